// hhgnnConv_eu_without_hetero_26259430048774
// MI455X (gfx1250) — compile-verified
//
#include <hip/hip_runtime.h>
#include <math.h>

typedef __attribute__((ext_vector_type(2))) float v2f;
typedef __attribute__((ext_vector_type(8))) float v8f;

// ---------------------------------------------------------------------------
// Dense GEMM: X0 = X(N x 128) @ W(128 x 128) + b, fp32 WMMA (16x16x4).
// Block = 256 threads (8 waves). Block tile: 64 rows x 128 cols.
// Wave w owns columns [16w, 16w+16) and 4 row-tiles of 16 rows each.
// ---------------------------------------------------------------------------
static __device__ __forceinline__ v8f wmma_f32_k4(v2f a, v2f b, v8f c) {
  // D = A(16x4 f32) * B(4x16 f32) + C(16x16 f32)
  return __builtin_amdgcn_wmma_f32_16x16x4_f32(
      /*neg_a=*/false, a, /*neg_b=*/false, b,
      /*c_mod=*/(short)0, c, /*reuse_a=*/false, /*reuse_b=*/false);
}

__global__ __launch_bounds__(256) void gemm_bias_kernel(
    const float* __restrict__ X, const float* __restrict__ W,
    const float* __restrict__ bias, float* __restrict__ X0, int N) {
  __shared__ float Xs[64][132];   // 64 rows x 128 K (pad 132: conflict-free)
  __shared__ float Ws[32][136];   // 32 K x 128 cols (pad 136: conflict-free)
  const int tid = threadIdx.x;
  const int wave = tid >> 5;
  const int lane = tid & 31;
  const int row0 = blockIdx.x * 64;

  for (int i = tid; i < 64 * 128; i += 256) {
    int r = i >> 7, c = i & 127;
    int gr = row0 + r;
    Xs[r][c] = (gr < N) ? X[(size_t)gr * 128 + c] : 0.0f;
  }

  // A-frag (16x4 f32): lane m=lane&15 holds K=koff,koff+1 (koff=0 lanes 0-15, 2 lanes 16-31)
  const int m    = lane & 15;
  const int koff = (lane >> 4) << 1;
  // B-frag (4x16 f32): lane holds col n=lane&15, rows K=koff,koff+1
  const int ncol = (wave << 4) + (lane & 15);

  v8f acc0 = {}, acc1 = {}, acc2 = {}, acc3 = {};

  for (int kk = 0; kk < 128; kk += 32) {
    __syncthreads();                       // Xs ready / Ws reuse safe
    for (int i = tid; i < 32 * 128; i += 256) {
      int r = i >> 7, c = i & 127;
      Ws[r][c] = W[(size_t)(kk + r) * 128 + c];
    }
    __syncthreads();
#pragma unroll
    for (int k = 0; k < 32; k += 4) {
      v2f b;
      b.x = Ws[k + koff][ncol];
      b.y = Ws[k + koff + 1][ncol];
      const int gk = kk + k + koff;
      v2f a;
      a.x = Xs[m][gk];      a.y = Xs[m][gk + 1];      acc0 = wmma_f32_k4(a, b, acc0);
      a.x = Xs[16 + m][gk]; a.y = Xs[16 + m][gk + 1]; acc1 = wmma_f32_k4(a, b, acc1);
      a.x = Xs[32 + m][gk]; a.y = Xs[32 + m][gk + 1]; acc2 = wmma_f32_k4(a, b, acc2);
      a.x = Xs[48 + m][gk]; a.y = Xs[48 + m][gk + 1]; acc3 = wmma_f32_k4(a, b, acc3);
    }
  }

  // C/D layout: VGPR r -> M = r + (lane>=16 ? 8 : 0), N = lane&15 within tile
  const float bv    = bias[ncol];
  const int   mbase = (lane >> 4) << 3;
#pragma unroll
  for (int r = 0; r < 8; ++r) {
    int lm = mbase + r, g;
    g = row0 + lm;      if (g < N) X0[(size_t)g * 128 + ncol] = acc0[r] + bv;
    g = row0 + 16 + lm; if (g < N) X0[(size_t)g * 128 + ncol] = acc1[r] + bv;
    g = row0 + 32 + lm; if (g < N) X0[(size_t)g * 128 + ncol] = acc2[r] + bv;
    g = row0 + 48 + lm; if (g < N) X0[(size_t)g * 128 + ncol] = acc3[r] + bv;
  }
}

// ---------------------------------------------------------------------------
// Helpers / elementwise kernels
// ---------------------------------------------------------------------------
static __device__ __forceinline__ void atomicMaxF(float* addr, float v) {
  // sign-aware int trick: works with -inf init
  if (v >= 0.0f) atomicMax((int*)addr, __float_as_int(v));
  else           atomicMin((unsigned int*)addr, __float_as_uint(v));
}

__global__ void fill_f32(float* p, long long n, float v) {
  long long i = (long long)blockIdx.x * blockDim.x + threadIdx.x;
  long long s = (long long)gridDim.x * blockDim.x;
  for (; i < n; i += s) p[i] = v;
}

// M-sized fills: M (edge_num) lives in a device scalar -> grid-stride, bound on device
__global__ void fill_f32_dyn(float* p, const int* __restrict__ mPtr, int mult, float v) {
  long long n = (long long)(*mPtr) * mult;
  long long i = (long long)blockIdx.x * blockDim.x + threadIdx.x;
  long long s = (long long)gridDim.x * blockDim.x;
  for (; i < n; i += s) p[i] = v;
}

__global__ void relu_dyn(float* p, const int* __restrict__ mPtr) {
  long long n = (long long)(*mPtr) * 128;
  long long i = (long long)blockIdx.x * blockDim.x + threadIdx.x;
  long long s = (long long)gridDim.x * blockDim.x;
  for (; i < n; i += s) p[i] = fmaxf(p[i], 0.0f);
}

// ---------------------------------------------------------------------------
// Phase kernels, shared by both attention passes. One thread per (i, head).
// ---------------------------------------------------------------------------
__global__ void score_max_kernel(const float* __restrict__ F, const int* __restrict__ srcIdx,
                                 const int* __restrict__ segIdx, const float* __restrict__ att,
                                 float* __restrict__ score, float* __restrict__ segmax, int nnz) {
  int t = blockIdx.x * blockDim.x + threadIdx.x;
  if (t >= nnz * 8) return;
  int i = t >> 3, h = t & 7;
  const float4* f = (const float4*)(F + (size_t)srcIdx[i] * 128 + h * 16);
  const float4* a = (const float4*)(att + h * 16);
  float s = 0.0f;
#pragma unroll
  for (int c = 0; c < 4; ++c) {
    float4 fv = f[c], av = a[c];
    s += fv.x * av.x + fv.y * av.y + fv.z * av.z + fv.w * av.w;
  }
  s = (s > 0.0f) ? s : 0.2f * s;           // leaky_relu, NEG_SLOPE=0.2
  score[t] = s;
  atomicMaxF(&segmax[segIdx[i] * 8 + h], s);
}

__global__ void exp_sum_kernel(float* __restrict__ score, const int* __restrict__ segIdx,
                               const float* __restrict__ segmax, float* __restrict__ segsum,
                               int nnz) {
  int t = blockIdx.x * blockDim.x + threadIdx.x;
  if (t >= nnz * 8) return;
  int i = t >> 3, h = t & 7;
  int sg = segIdx[i] * 8 + h;
  float m = segmax[sg];
  if (m < -3.0e38f) m = 0.0f;              // reference: where(isfinite(m), m, 0)
  float e = expf(score[t] - m);
  score[t] = e;
  atomicAdd(&segsum[sg], e);
}

__global__ void scatter_weighted_kernel(const float* __restrict__ F, const int* __restrict__ srcIdx,
                                        const int* __restrict__ segIdx, const float* __restrict__ e,
                                        const float* __restrict__ segsum, float* __restrict__ outF,
                                        int nnz) {
  int t = blockIdx.x * blockDim.x + threadIdx.x;
  if (t >= nnz * 8) return;
  int i = t >> 3, h = t & 7;
  int sg = segIdx[i];
  float w = e[t] / (segsum[(size_t)sg * 8 + h] + 1e-16f);
  const float* f = F + (size_t)srcIdx[i] * 128 + h * 16;
  float*       o = outF + (size_t)sg * 128 + h * 16;
#pragma unroll
  for (int c = 0; c < 16; ++c) atomicAdd(o + c, f[c] * w);
}

// ---------------------------------------------------------------------------
// Launch
// ---------------------------------------------------------------------------
extern "C" void kernel_launch(void* const* d_in, const int* in_sizes, int n_in,
                              void* d_out, int out_size, void* d_ws, size_t ws_size,
                              hipStream_t stream) {
  const float* X      = (const float*)d_in[0];
  const int*   vertex = (const int*)d_in[1];
  const int*   edges  = (const int*)d_in[2];
  const float* W_w    = (const float*)d_in[3];
  const float* W_b    = (const float*)d_in[4];
  const float* att_e  = (const float*)d_in[5];
  const float* att_v  = (const float*)d_in[6];
  // inputs 7..16 (class partitions) are a mathematical identity -> unused
  const int*   edge_num = (const int*)d_in[17];   // device scalar M

  const int N   = in_sizes[0] / 128;
  const int NNZ = in_sizes[1];
  float* out = (float*)d_out;

  // workspace layout (fp32); M-sized buffers use capacity N (M <= N here)
  float* ws = (float*)d_ws;
  size_t off = 0;
  float* X0     = ws + off; off += (size_t)N * 128;   // X @ W + b
  float* Xe     = ws + off; off += (size_t)N * 128;   // per-edge features (cap N)
  float* score  = ws + off; off += (size_t)NNZ * 8;   // beta/alpha scratch
  float* segmax = ws + off; off += (size_t)N * 8;
  float* segsum = ws + off; off += (size_t)N * 8;
  (void)n_in; (void)ws_size;

  const int TB = 256;
  dim3 blk(TB);
  const int fillBlocks  = 2048;
  const int scoreBlocks = (NNZ * 8 + TB - 1) / TB;

  // 1) dense GEMM via fp32 WMMA
  gemm_bias_kernel<<<(N + 63) / 64, blk, 0, stream>>>(X, W_w, W_b, X0, N);

  // 2) edge-side attention + softmax + scatter into Xe (segment = edges, size M)
  fill_f32_dyn<<<fillBlocks, blk, 0, stream>>>(segmax, edge_num, 8,  -INFINITY);
  fill_f32_dyn<<<fillBlocks, blk, 0, stream>>>(segsum, edge_num, 8,   0.0f);
  fill_f32_dyn<<<fillBlocks, blk, 0, stream>>>(Xe,     edge_num, 128, 0.0f);
  score_max_kernel       <<<scoreBlocks, blk, 0, stream>>>(X0, vertex, edges, att_e, score, segmax, NNZ);
  exp_sum_kernel         <<<scoreBlocks, blk, 0, stream>>>(score, edges, segmax, segsum, NNZ);
  scatter_weighted_kernel<<<scoreBlocks, blk, 0, stream>>>(X0, vertex, edges, score, segsum, Xe, NNZ);
  relu_dyn               <<<fillBlocks,  blk, 0, stream>>>(Xe, edge_num);

  // 3) vertex-side attention + softmax + scatter into out (segment = vertex, size N)
  fill_f32<<<fillBlocks, blk, 0, stream>>>(segmax, (long long)N * 8, -INFINITY);
  fill_f32<<<fillBlocks, blk, 0, stream>>>(segsum, (long long)N * 8, 0.0f);
  fill_f32<<<fillBlocks, blk, 0, stream>>>(out,    (long long)out_size, 0.0f);
  score_max_kernel       <<<scoreBlocks, blk, 0, stream>>>(Xe, edges, vertex, att_v, score, segmax, NNZ);
  exp_sum_kernel         <<<scoreBlocks, blk, 0, stream>>>(score, vertex, segmax, segsum, NNZ);
  scatter_weighted_kernel<<<scoreBlocks, blk, 0, stream>>>(Xe, edges, vertex, score, segsum, out, NNZ);
}